// Cumulative_Probability_Layer_37615323579135
// MI455X (gfx1250) — compile-verified
//
#include <hip/hip_runtime.h>
#include <stdint.h>

typedef __attribute__((ext_vector_type(2))) float v2f;
typedef __attribute__((ext_vector_type(8))) float v8f;

#define D_FEAT        1024
#define T_FOLLOW      6
#define OUT_COLS      7
#define ROWS_PER_WAVE 16
#define WAVES_PER_WG  8
#define ROWS_PER_WG   (ROWS_PER_WAVE * WAVES_PER_WG)   // 128
#define K_CHUNK       64
#define N_CHUNKS      (D_FEAT / K_CHUNK)               // 16
#define XS_STRIDE     68         // pad: compute-side bank = (4*row + col) % 64, conflict-free
#define WCT_ROWS      8          // cols 0..6 = weights, row 7 = zeros (N>=7 alias here)
#define WCT_STRIDE    1026       // pad: bank = (2n + k) % 64, conflict-free per half-wave
#define TILE_STRIDE   17         // padded 16x16 epilogue transpose tile

__global__ __launch_bounds__(256) void hazard_head_wmma_async_kernel(
    const float* __restrict__ x,         // [32768, 1024]
    const float* __restrict__ W_hazard,  // [1024, 6]
    const float* __restrict__ b_hazard,  // [6]
    const float* __restrict__ W_base,    // [1024, 1]
    const float* __restrict__ b_base,    // [1]
    float* __restrict__ out)             // [32768, 7]
{
    __shared__ float xs[2][ROWS_PER_WG * XS_STRIDE];     // double-buffered A staging
    __shared__ float wcT[WCT_ROWS * WCT_STRIDE];         // combined transposed weights
    __shared__ float tile[WAVES_PER_WG][ROWS_PER_WAVE * TILE_STRIDE];

    const int tid = threadIdx.x;

    // Build combined transposed weight matrix in LDS:
    //   wcT[n][k]: n==0 -> W_base[k], n in 1..6 -> W_hazard[k][n-1], n==7 -> 0
    #pragma unroll
    for (int i = 0; i < (WCT_ROWS * D_FEAT) / 256; ++i) {
        int idx = i * 256 + tid;
        int n = idx >> 10;
        int k = idx & (D_FEAT - 1);
        float v = 0.0f;
        if (n == 0)             v = W_base[k];
        else if (n <= T_FOLLOW) v = W_hazard[k * T_FOLLOW + (n - 1)];
        wcT[n * WCT_STRIDE + k] = v;
    }

    // ---- async staging setup: thread t owns (row = r*16 + t>>4, 4 floats at col (t&15)*4)
    const int srow = tid >> 4;                  // 0..15
    const int scol = (tid & 15) * 4;            // 0..60
    const int rowBaseWG = blockIdx.x * ROWS_PER_WG;
    const float* gthread = x + (size_t)(rowBaseWG + srow) * D_FEAT + scol;
    const uint32_t lthread0 = (uint32_t)(uintptr_t)&xs[0][srow * XS_STRIDE + scol];
    const uint32_t lthread1 = (uint32_t)(uintptr_t)&xs[1][srow * XS_STRIDE + scol];

    // issue one K-chunk (128 rows x 64 cols) as 8 coalesced async b128 loads/thread
    auto issue_chunk = [&](int c, int buf) {
        const float* g = gthread + c * K_CHUNK;
        uint32_t l = buf ? lthread1 : lthread0;
        #pragma unroll
        for (int r = 0; r < 8; ++r) {
            const float* ga = g + r * 16 * D_FEAT;                  // 16 rows apart
            uint32_t la = l + (uint32_t)(r * 16 * XS_STRIDE * 4);
            asm volatile("global_load_async_to_lds_b128 %0, %1, off"
                         :: "v"(la), "v"(ga) : "memory");
        }
    };

    issue_chunk(0, 0);          // prime the pipeline

    const int wave = tid >> 5;
    const int lane = tid & 31;
    const int half = lane >> 4;     // fp32 A/B layout: half 0 -> K+0/K+1, half 1 -> K+2/K+3
    const int l16  = lane & 15;

    // compute-side LDS pointers
    const int bRow = (l16 < OUT_COLS) ? l16 : (WCT_ROWS - 1);       // zeros row for N>=7
    const float* bp  = &wcT[bRow * WCT_STRIDE + 2 * half];
    const float* ap0 = &xs[0][(wave * ROWS_PER_WAVE + l16) * XS_STRIDE + 2 * half];
    const float* ap1 = &xs[1][(wave * ROWS_PER_WAVE + l16) * XS_STRIDE + 2 * half];

    v8f c = {0.f, 0.f, 0.f, 0.f, 0.f, 0.f, 0.f, 0.f};

    for (int ch = 0; ch < N_CHUNKS; ++ch) {
        asm volatile("s_wait_asynccnt 0" ::: "memory");   // chunk ch resident in LDS
        __syncthreads();                                  // all waves' staging visible
        if (ch + 1 < N_CHUNKS)
            issue_chunk(ch + 1, (ch + 1) & 1);            // overlap next chunk with compute

        const float* ap = (ch & 1) ? ap1 : ap0;
        const float* bc = bp + ch * K_CHUNK;
        #pragma unroll
        for (int k = 0; k < K_CHUNK; k += 4) {
            v2f a = *(const v2f*)(ap + k);                // ds_load_b64 (conflict-free)
            v2f b = *(const v2f*)(bc + k);                // ds_load_b64 (broadcast for N>=7)
            c = __builtin_amdgcn_wmma_f32_16x16x4_f32(
                    false, a, false, b, (short)0, c, false, false);
        }
    }

    // Epilogue: transpose this wave's 16x16 C tile through LDS so each row's
    // 7 columns land in one lane, then fuse bias + ReLU + inclusive cumsum.
    float* myTile = tile[wave];
    #pragma unroll
    for (int v = 0; v < 8; ++v)
        myTile[(v + 8 * half) * TILE_STRIDE + l16] = c[v];   // M = v + 8*half, N = l16
    __builtin_amdgcn_wave_barrier();

    if (lane < 16) {
        const int row = rowBaseWG + wave * ROWS_PER_WAVE + lane;
        float acc = myTile[lane * TILE_STRIDE + 0] + b_base[0];   // base hazard
        float* o = out + (size_t)row * OUT_COLS;
        o[0] = acc;
        #pragma unroll
        for (int j = 1; j <= T_FOLLOW; ++j) {
            float h = myTile[lane * TILE_STRIDE + j] + b_hazard[j - 1];
            h = fmaxf(h, 0.0f);                                   // ReLU
            acc += h;                                             // inclusive cumsum + base
            o[j] = acc;
        }
    }
}

extern "C" void kernel_launch(void* const* d_in, const int* in_sizes, int n_in,
                              void* d_out, int out_size, void* d_ws, size_t ws_size,
                              hipStream_t stream) {
    (void)in_sizes; (void)n_in; (void)out_size; (void)d_ws; (void)ws_size;
    const float* x        = (const float*)d_in[0];
    const float* W_hazard = (const float*)d_in[1];
    const float* b_hazard = (const float*)d_in[2];
    const float* W_base   = (const float*)d_in[3];
    const float* b_base   = (const float*)d_in[4];
    float* out = (float*)d_out;

    const int n_rows = 32768;
    dim3 grid(n_rows / ROWS_PER_WG);   // 256 workgroups
    dim3 block(256);                   // 8 wave32s
    hazard_head_wmma_async_kernel<<<grid, block, 0, stream>>>(
        x, W_hazard, b_hazard, W_base, b_base, out);
}